// SovitsV5B_66821101191272
// MI455X (gfx1250) — compile-verified
//
#include <hip/hip_runtime.h>
#include <math.h>

// ---------------- problem constants ----------------
#define SRATE   44100
#define BLK     512
#define BATCH   8
#define NFRM    2048                 // N
#define NBIN    513
#define TLEN    (NFRM * BLK)         // 1048576
#define PI_F    3.14159265358979323846f

typedef float v2f __attribute__((ext_vector_type(2)));
typedef float v8f __attribute__((ext_vector_type(8)));

// ---------------------------------------------------------------------------
// Kernel A: exclusive per-block phase bases (closed-form block sums + scan).
// ---------------------------------------------------------------------------
__global__ __launch_bounds__(256)
void k_phase_scan(const float* __restrict__ f0f, double* __restrict__ base)
{
    const int b = blockIdx.x;       // 0..7
    const int t = threadIdx.x;      // 0..255
    double loc[8];
    double tot = 0.0;
#pragma unroll
    for (int i = 0; i < 8; ++i) {
        int j = t * 8 + i;
        float fa = f0f[b * NFRM + j];
        float fb = (j + 1 < NFRM) ? f0f[b * NFRM + j + 1] : f0f[b * NFRM + NFRM - 1];
        double s = (256.5 * (double)fa + 255.5 * (double)fb) * (1.0 / (double)SRATE);
        loc[i] = tot;               // exclusive within-thread
        tot += s;
    }
    __shared__ double sc[256];
    sc[t] = tot;
    __syncthreads();
    for (int off = 1; off < 256; off <<= 1) {
        double v = (t >= off) ? sc[t - off] : 0.0;
        __syncthreads();
        sc[t] += v;
        __syncthreads();
    }
    double excl = (t == 0) ? 0.0 : sc[t - 1];
#pragma unroll
    for (int i = 0; i < 8; ++i)
        base[b * NFRM + t * 8 + i] = excl + loc[i];
}

// ---------------------------------------------------------------------------
// CDNA5 async global -> LDS copy (ASYNCcnt-tracked), per-lane addresses.
//   lds_byte : LDS byte address (VGPR), gbase : uniform SGPR base,
//   voff     : per-lane byte offset (GVS mode).
// ---------------------------------------------------------------------------
__device__ __forceinline__
void async_g2l_b32(unsigned lds_byte, const float* gbase, unsigned voff)
{
    asm volatile("global_load_async_to_lds_b32 %0, %1, %2"
                 :
                 : "v"(lds_byte), "v"(voff), "s"(gbase)
                 : "memory");
}
__device__ __forceinline__ void wait_async(int n)
{
    if (n == 0)       asm volatile("s_wait_asynccnt 0"  ::: "memory");
    else if (n == 12) asm volatile("s_wait_asynccnt 12" ::: "memory");
}
__device__ __forceinline__ unsigned lds_addr(const void* p)
{
    // generic LDS pointer: low 32 bits are the workgroup-relative LDS address
    return (unsigned)(uintptr_t)p;
}

// ---------------------------------------------------------------------------
// WMMA helper: D(16x16,f32) = A(16x4) * B(4x16) + C
// ---------------------------------------------------------------------------
__device__ __forceinline__ v8f wmma4(v2f a, v2f b, v8f c)
{
    return __builtin_amdgcn_wmma_f32_16x16x4_f32(false, a, false, b,
                                                 (short)0, c, false, false);
}

__device__ __forceinline__ int brev6(int x)
{
    return (int)(__brev((unsigned)x) >> 26);
}

// ---------------------------------------------------------------------------
// Forward 1024-pt complex FFT, four-step 16x64:
//   stage1: A = DFT16 * Z          (WMMA f32 16x16x4, 4 waves x 4 col-tiles)
//   stage2: twiddle W1024^{n2*k1}  (chained complex multiply in C-fragments)
//   stage3: 64-pt radix-2 DIF per row, twiddles from 32-entry LDS table
// Final X[k1 + 16*k2] lives at br/bi[64*k1 + bitrev6(k2)].
// Caller must __syncthreads() before calling (zr/zi fully written).
// ---------------------------------------------------------------------------
__device__ __forceinline__
void fft1024(float* zr, float* zi, float* br, float* bi,
             const float* twc, const float* tws,
             const v2f* Dr, const v2f* Di, const v2f* Dni,
             int lane, int wave)
{
    const int m  = lane & 15;
    const int th = lane >> 4;
    const int n2 = wave * 16 + m;

    v8f aR = {0.f,0.f,0.f,0.f,0.f,0.f,0.f,0.f};
    v8f aI = {0.f,0.f,0.f,0.f,0.f,0.f,0.f,0.f};
#pragma unroll
    for (int c = 0; c < 4; ++c) {
        int k0 = 4 * c + 2 * th;                 // ISA A/B fragment K layout
        v2f bzr, bzi;
        bzr.x = zr[k0 * 64 + n2];  bzr.y = zr[(k0 + 1) * 64 + n2];
        bzi.x = zi[k0 * 64 + n2];  bzi.y = zi[(k0 + 1) * 64 + n2];
        aR = wmma4(Dr[c],  bzr, aR);             // Re += Dr*Zr
        aR = wmma4(Dni[c], bzi, aR);             // Re += (-Di)*Zi
        aI = wmma4(Dr[c],  bzi, aI);             // Im += Dr*Zi
        aI = wmma4(Di[c],  bzr, aI);             // Im += Di*Zr
    }
    // stage2: B = A * e^{-i*theta}, theta = 2pi*n2*(r+8*th)/1024.
    // Geometric in r: 2 sincos + chained complex multiply.
    {
        float d  = (2.0f * PI_F / 1024.0f) * (float)n2;
        float sd, cd; __sincosf(d, &sd, &cd);
        float s0, c0; __sincosf(d * (float)(8 * th), &s0, &c0);
#pragma unroll
        for (int r = 0; r < 8; ++r) {
            int   k1 = r + 8 * th;
            float Ar = aR[r], Ai = aI[r];
            br[k1 * 64 + n2] = Ar * c0 + Ai * s0;
            bi[k1 * 64 + n2] = Ai * c0 - Ar * s0;
            float cn = c0 * cd - s0 * sd;        // advance twiddle
            s0 = s0 * cd + c0 * sd;
            c0 = cn;
        }
    }
    // stage3: radix-2 DIF, 6 stages; twiddle e^{-i*pi*pos/half} = T[pos<<(5-s)]
    for (int s = 5; s >= 0; --s) {
        __syncthreads();
        const int half = 1 << s;
        for (int g = threadIdx.x; g < 512; g += 128) {
            int row = g >> 5, j = g & 31;
            int pos = j & (half - 1);
            int i0  = row * 64 + ((j >> s) << (s + 1)) + pos;
            int i1  = i0 + half;
            float ur = br[i0], ui = bi[i0];
            float vr = br[i1], vi = bi[i1];
            br[i0] = ur + vr;  bi[i0] = ui + vi;
            float dr = ur - vr, di = ui - vi;
            int   u  = pos << (5 - s);
            float cs = twc[u], sn = tws[u];
            br[i1] = dr * cs + di * sn;
            bi[i1] = di * cs - dr * sn;
        }
    }
    __syncthreads();
}

// ---------------------------------------------------------------------------
// Kernel B: one workgroup (128 thr = 4 wave32) per (batch, frame k), k=0..N.
// ---------------------------------------------------------------------------
__global__ __launch_bounds__(128)
void k_frames(const float* __restrict__ f0f,
              const float* __restrict__ hm,
              const float* __restrict__ hp,
              const float* __restrict__ nm,
              const float* __restrict__ noise,
              const double* __restrict__ base,
              float* __restrict__ out)
{
    __shared__ float zr[1024], zi[1024], br[1024], bi[1024];
    __shared__ float sh_hm[512], sh_hp[512], sh_nm[512];
    __shared__ float sh_no[1024];
    __shared__ float twc[32], tws[32];

    const int b    = blockIdx.x / (NFRM + 1);
    const int k    = blockIdx.x % (NFRM + 1);
    const int tid  = threadIdx.x;
    const int lane = tid & 31;
    const int wave = tid >> 5;
    const int m    = lane & 15;
    const int th   = lane >> 4;

    const int    kf = (k < NFRM) ? k : (NFRM - 1);
    const size_t sb = ((size_t)b * NFRM + kf) * NBIN;

    // ---- issue async prefetches first: noise frame (8 instr/wave), then
    //      spectral arrays for bins 0..511 (12 instr/wave, uniform) ----------
    const float* noise_b = noise + (size_t)b * TLEN;
#pragma unroll
    for (int n = tid; n < 1024; n += 128) {
        long s = (long)(k - 1) * BLK + n;
        if (s >= 0 && s < (long)TLEN)
            async_g2l_b32(lds_addr(&sh_no[n]), noise_b, (unsigned)(s * 4));
    }
    const float* hmb = hm + sb;
    const float* hpb = hp + sb;
    const float* nmb = nm + sb;
#pragma unroll
    for (int kb = tid; kb < 512; kb += 128) {
        unsigned vo = (unsigned)(kb * 4);
        async_g2l_b32(lds_addr(&sh_hm[kb]), hmb, vo);
        async_g2l_b32(lds_addr(&sh_hp[kb]), hpb, vo);
        async_g2l_b32(lds_addr(&sh_nm[kb]), nmb, vo);
    }

    // ---- DFT16 fragments: Dr=cos(2pi*m*kk/16), Di=-sin, Dni=+sin ----------
    v2f Dr[4], Di[4], Dni[4];
#pragma unroll
    for (int c = 0; c < 4; ++c) {
        int k0 = 4 * c + 2 * th;
        float a0 = (PI_F / 8.0f) * (float)((m * k0) & 15);
        float a1 = (PI_F / 8.0f) * (float)((m * (k0 + 1)) & 15);
        float s0, c0, s1, c1;
        __sincosf(a0, &s0, &c0);
        __sincosf(a1, &s1, &c1);
        Dr[c].x  = c0;  Dr[c].y  = c1;
        Di[c].x  = -s0; Di[c].y  = -s1;
        Dni[c].x = s0;  Dni[c].y = s1;
    }
    // ---- DIF twiddle table T[t] = e^{-i*pi*t/32} ---------------------------
    if (tid < 32) {
        float sv, cv; __sincosf((PI_F / 32.0f) * (float)tid, &sv, &cv);
        twc[tid] = cv;  tws[tid] = sv;
    }

    // ---- combtooth (on the fly, double closed-form phase) -> zr -----------
    for (int n = tid; n < 1024; n += 128) {
        float w  = __sinf(PI_F * (float)n * (1.0f / 1024.0f));  // sqrt-Hann
        long  s  = (long)(k - 1) * BLK + n;
        float cv = 0.f;
        if (s >= 0 && s < (long)TLEN) {
            int j = (int)(s >> 9), i = (int)(s & 511);
            float fa = f0f[b * NFRM + j];
            float fb = (j + 1 < NFRM) ? f0f[b * NFRM + j + 1]
                                      : f0f[b * NFRM + NFRM - 1];
            float fi = (float)i * (1.0f / 512.0f);
            float f0 = fa * (1.0f - fi) + fb * fi;
            double P = ((double)(i + 1) * (double)fa
                       + ((double)fb - (double)fa) * ((double)(i * (i + 1)) * (0.5 / 512.0)))
                       * (1.0 / (double)SRATE);
            double x  = base[b * NFRM + j] + P;
            double fr = x - rint(x);
            float  z  = (float)SRATE * (float)fr / (f0 + 0.001f);
            float  pz = PI_F * z;
            cv = (fabsf(pz) < 1e-8f) ? 1.0f : (sinf(pz) / pz);
        }
        zr[n] = cv * w;
    }

    // ---- noise arrived (first 8 async per wave): window into zi -----------
    wait_async(12);
    for (int n = tid; n < 1024; n += 128) {
        float w = __sinf(PI_F * (float)n * (1.0f / 1024.0f));
        long  s = (long)(k - 1) * BLK + n;
        float nv = (s >= 0 && s < (long)TLEN) ? (sh_no[n] * 2.0f - 1.0f) : 0.f;
        zi[n] = nv * w;
    }
    __syncthreads();

    fft1024(zr, zi, br, bi, twc, tws, Dr, Di, Dni, lane, wave);

    // ---- spectral combine (bins 0..512), write conj(Y) Hermitian-extended -
    wait_async(0);                      // spectral prefetch complete
    for (int kb = tid; kb < NBIN; kb += 128) {
        int i1 = ((kb & 15) << 6) | brev6(kb >> 4);
        int km = (1024 - kb) & 1023;
        int i2 = ((km & 15) << 6) | brev6(km >> 4);
        float z1r = br[i1], z1i = bi[i1];
        float z2r = br[i2], z2i = bi[i2];
        float Ar = 0.5f * (z1r + z2r), Ai = 0.5f * (z1i - z2i);   // comb_fft
        float Nr = 0.5f * (z1i + z2i), Ni = 0.5f * (z2r - z1r);   // noise_fft
        float vh = (kb < 512) ? sh_hm[kb] : hm[sb + 512];
        float vp = (kb < 512) ? sh_hp[kb] : hp[sb + 512];
        float vn = (kb < 512) ? sh_nm[kb] : nm[sb + 512];
        float eh = __expf(vh);
        float sp, cp; __sincosf(PI_F * vp, &sp, &cp);
        float sr = eh * cp, si = eh * sp;                          // src
        float nf = __expf(vn) * (1.0f / 128.0f);                   // nfil
        float Yr = Ar * sr - Ai * si + Nr * nf;
        float Yi = Ar * si + Ai * sr + Ni * nf;
        zr[kb] = Yr;  zi[kb] = -Yi;                    // conj(Y[k])
        if (kb >= 1 && kb <= 511) {                    // mirror: conj twice
            zr[1024 - kb] = Yr;  zi[1024 - kb] = Yi;
        }
    }
    __syncthreads();

    fft1024(zr, zi, br, bi, twc, tws, Dr, Di, Dni, lane, wave);

    // ---- Re(.)/1024 * win, overlap-add (exactly 2 adds/sample) ------------
    for (int n = tid; n < 1024; n += 128) {
        int   idx = ((n & 15) << 6) | brev6(n >> 4);
        float w   = __sinf(PI_F * (float)n * (1.0f / 1024.0f));
        float val = br[idx] * (1.0f / 1024.0f) * w;
        if (n < BLK) {
            if (k >= 1)
                atomicAdd(&out[(size_t)b * TLEN + (size_t)(k - 1) * BLK + n], val);
        } else {
            if (k <= NFRM - 1)
                atomicAdd(&out[(size_t)b * TLEN + (size_t)k * BLK + (n - BLK)], val);
        }
    }
}

// ---------------------------------------------------------------------------
extern "C" void kernel_launch(void* const* d_in, const int* in_sizes, int n_in,
                              void* d_out, int out_size, void* d_ws, size_t ws_size,
                              hipStream_t stream)
{
    const float* f0f   = (const float*)d_in[0];
    const float* hmag  = (const float*)d_in[1];
    const float* hph   = (const float*)d_in[2];
    const float* nmag  = (const float*)d_in[3];
    const float* noise = (const float*)d_in[4];
    float*  out  = (float*)d_out;
    double* base = (double*)d_ws;               // B*N doubles = 128 KB

    hipMemsetAsync(out, 0, sizeof(float) * (size_t)out_size, stream);
    k_phase_scan<<<BATCH, 256, 0, stream>>>(f0f, base);
    k_frames<<<BATCH * (NFRM + 1), 128, 0, stream>>>(f0f, hmag, hph, nmag,
                                                     noise, base, out);
}